// VectorQuantizerEMA_31482110279967
// MI455X (gfx1250) — compile-verified
//
#include <hip/hip_runtime.h>
#include <hip/hip_bf16.h>
#include <math.h>

typedef __attribute__((ext_vector_type(2))) float v2f;
typedef __attribute__((ext_vector_type(8))) float v8f;

#define EMB_K 512
#define EMB_D 64
#define ROWS_PER_BLOCK 128      // 8 waves * 16 rows
#define CODES_PER_CHUNK 128
#define LDS_STRIDE 66           // 64 + 2 pad -> conflict-free column reads

// ---------------------------------------------------------------------------
// Kernel 1: per-code squared norms + zero the histogram
// ---------------------------------------------------------------------------
__global__ __launch_bounds__(256) void vq_init_kernel(
    const float* __restrict__ emb, float* __restrict__ enorms,
    int* __restrict__ counts) {
  int k = blockIdx.x * 256 + threadIdx.x;
  if (k < EMB_K) {
    float s = 0.f;
    const float* e = emb + (long)k * EMB_D;
#pragma unroll 8
    for (int d = 0; d < EMB_D; ++d) { float v = e[d]; s += v * v; }
    enorms[k] = s;
    counts[k] = 0;
  }
}

// ---------------------------------------------------------------------------
// Kernel 2: WMMA distance GEMM + argmin + gather + partial SSE
// ---------------------------------------------------------------------------
__global__ __launch_bounds__(256) void vq_main_kernel(
    const float* __restrict__ x, const float* __restrict__ emb,
    const float* __restrict__ enorms, float* __restrict__ out,
    int* __restrict__ counts, float* __restrict__ bsse, int nrows, int nflat) {
  __shared__ float elds[CODES_PER_CHUNK * LDS_STRIDE];  // ~33.8 KB
  __shared__ int   sidx[ROWS_PER_BLOCK];
  __shared__ float wsum[8];

  const int tid  = threadIdx.x;
  const int wave = tid >> 5;
  const int lane = tid & 31;
  const int half = lane >> 4;   // 0: rows/K 0..; 1: upper half per ISA layout
  const int lh   = lane & 15;

  const long rowbase = (long)blockIdx.x * ROWS_PER_BLOCK + wave * 16;

  // --- A fragments: 16 k-steps of the 16x4 f32 A tile, kept in VGPRs ---
  // Lane layout (ISA 7.12.2, 32-bit A 16x4): lanes 0-15 hold M=0..15 with
  // K={0,1} in the VGPR pair; lanes 16-31 hold K={2,3}.
  v2f a[16];
  {
    const float* xr = x + (rowbase + lh) * EMB_D + half * 2;
#pragma unroll
    for (int s = 0; s < 16; ++s) {
      float2 t = *(const float2*)(xr + s * 4);
      a[s].x = t.x; a[s].y = t.y;
    }
  }

  // Running per-lane (min, argmin) in C-matrix slots.
  float minv[8];
  int   mini[8];
#pragma unroll
  for (int j = 0; j < 8; ++j) { minv[j] = 3.4e38f; mini[j] = 0; }

  for (int chunk = 0; chunk < EMB_K / CODES_PER_CHUNK; ++chunk) {
    __syncthreads();  // previous chunk fully consumed
    const float* src = emb + (long)chunk * CODES_PER_CHUNK * EMB_D;
    // stage 128 codes into LDS with padded stride
    for (int u = tid; u < CODES_PER_CHUNK * 16; u += 256) {
      int r = u >> 4, q = u & 15;
      float4 v = *(const float4*)(src + r * EMB_D + q * 4);
      float* d = &elds[r * LDS_STRIDE + q * 4];
      d[0] = v.x; d[1] = v.y; d[2] = v.z; d[3] = v.w;
    }
    if (chunk + 1 < EMB_K / CODES_PER_CHUNK)  // global_prefetch_b8 next chunk
      __builtin_prefetch(src + CODES_PER_CHUNK * EMB_D + tid * 64, 0, 0);
    __syncthreads();

#pragma unroll
    for (int ct = 0; ct < CODES_PER_CHUNK / 16; ++ct) {
      const int codebase = chunk * CODES_PER_CHUNK + ct * 16;
      v8f acc = {0.f, 0.f, 0.f, 0.f, 0.f, 0.f, 0.f, 0.f};
#pragma unroll
      for (int s = 0; s < 16; ++s) {
        // B tile 4x16: lane lh = column (code), VGPR pair = K{0,1}/{2,3}
        const float* bp = &elds[(ct * 16 + lh) * LDS_STRIDE + s * 4 + half * 2];
        v2f b; b.x = bp[0]; b.y = bp[1];
        acc = __builtin_amdgcn_wmma_f32_16x16x4_f32(
            /*neg_a=*/false, a[s], /*neg_b=*/false, b,
            /*c_mod=*/(short)0, acc, /*reuse_a=*/false, /*reuse_b=*/false);
      }
      const float en   = enorms[codebase + lh];
      const int   code = codebase + lh;
#pragma unroll
      for (int j = 0; j < 8; ++j) {
        float score = en - 2.0f * acc[j];  // ||e||^2 - 2 x.e (||x||^2 const)
        if (score < minv[j] || (score == minv[j] && code < mini[j])) {
          minv[j] = score; mini[j] = code;
        }
      }
    }
  }

  // Cross-lane argmin: C layout => row m lives on lanes 0-15 (m<8) or 16-31.
#pragma unroll
  for (int j = 0; j < 8; ++j) {
#pragma unroll
    for (int off = 8; off >= 1; off >>= 1) {
      float ov = __shfl_xor(minv[j], off, 16);
      int   oi = __shfl_xor(mini[j], off, 16);
      if (ov < minv[j] || (ov == minv[j] && oi < mini[j])) {
        minv[j] = ov; mini[j] = oi;
      }
    }
  }
  if (lh == 0) {
#pragma unroll
    for (int j = 0; j < 8; ++j) sidx[wave * 16 + half * 8 + j] = mini[j];
  }
  __syncthreads();

  // --- Output phase: gather codes, write quantized + indices, partial SSE ---
  float* qout   = out + 1;                 // quantized_st
  float* idxout = out + 2 + nflat;         // indices (after perplexity slot)
  float lsum = 0.f;
  for (int r = 0; r < 16; ++r) {
    const int  code = sidx[wave * 16 + r];
    const long row  = rowbase + r;
    float2 q  = *(const float2*)(emb + (long)code * EMB_D + lane * 2);
    float2 xx = *(const float2*)(x + row * EMB_D + lane * 2);
    *(float2*)(qout + row * EMB_D + lane * 2) = q;
    float dx = q.x - xx.x, dy = q.y - xx.y;
    lsum += dx * dx + dy * dy;
    if (lane == 0) {
      idxout[row] = (float)code;
      atomicAdd(&counts[code], 1);  // integer: deterministic
    }
  }
  // deterministic SSE: wave reduce -> block reduce -> per-block slot
#pragma unroll
  for (int off = 16; off >= 1; off >>= 1) lsum += __shfl_xor(lsum, off, 32);
  if (lane == 0) wsum[wave] = lsum;
  __syncthreads();
  if (tid == 0) {
    float s = 0.f;
#pragma unroll
    for (int w = 0; w < 8; ++w) s += wsum[w];
    bsse[blockIdx.x] = s;
  }
}

// ---------------------------------------------------------------------------
// Kernel 3: loss + perplexity (fixed-order reductions)
// ---------------------------------------------------------------------------
__global__ __launch_bounds__(512) void vq_fin_kernel(
    const int* __restrict__ counts, const float* __restrict__ bsse,
    float* __restrict__ out, int nblocks, int nrows, int nflat) {
  __shared__ float red[512];
  const int k = threadIdx.x;
  // entropy term
  float p = (float)counts[k] / (float)nrows;
  red[k] = p * logf(p + 1e-10f);
  __syncthreads();
  for (int s = 256; s > 0; s >>= 1) {
    if (k < s) red[k] += red[k + s];
    __syncthreads();
  }
  float H = red[0];
  __syncthreads();
  // SSE sum over blocks
  red[k] = (k < nblocks) ? bsse[k] : 0.f;
  __syncthreads();
  for (int s = 256; s > 0; s >>= 1) {
    if (k < s) red[k] += red[k + s];
    __syncthreads();
  }
  if (k == 0) {
    float mse = red[0] / (float)nflat;
    out[0]         = 1.25f * mse;   // q_latent + 0.25 * e_latent (equal fwd)
    out[1 + nflat] = expf(-H);      // perplexity
  }
}

// ---------------------------------------------------------------------------
extern "C" void kernel_launch(void* const* d_in, const int* in_sizes, int n_in,
                              void* d_out, int out_size, void* d_ws, size_t ws_size,
                              hipStream_t stream) {
  const float* x   = (const float*)d_in[0];  // [64,1024,64] f32
  const float* emb = (const float*)d_in[1];  // [512,64] f32
  float* out = (float*)d_out;
  float* ws  = (float*)d_ws;

  float* enorms = ws + 64;            // 512 floats
  float* bsse   = ws + 576;           // up to 512 floats
  int*   counts = (int*)(ws + 1088);  // 512 ints

  const int nflat   = in_sizes[0];        // N * D = 4194304
  const int nrows   = nflat / EMB_D;      // 65536
  const int nblocks = nrows / ROWS_PER_BLOCK;  // 512

  vq_init_kernel<<<2, 256, 0, stream>>>(emb, enorms, counts);
  vq_main_kernel<<<nblocks, 256, 0, stream>>>(x, emb, enorms, out, counts,
                                              bsse, nrows, nflat);
  vq_fin_kernel<<<1, 512, 0, stream>>>(counts, bsse, out, nblocks, nrows, nflat);
}